// GHME_Loss_70970039599228
// MI455X (gfx1250) — compile-verified
//
#include <hip/hip_runtime.h>

// GHM loss, single fused HBM pass (268 MB @ 23.3 TB/s ~ 11.5 us floor):
//   loss_total = sum_b beta_b * S_b,  S_b = sum of BCE over elements in bin b
// Pass 1 (ghm_main): per-thread LDS histograms (packed f32 sum + u32 count in u64),
//                    tree-reduce per WG, global atomics into d_ws.
// Pass 2 (ghm_finalize): one wave32 computes beta_b*S_b and reduces the 10 terms
//                        with v_wmma_f32_16x16x4_f32 (B = ones -> row sums).

#define BINS 10
#define TPB 256
#define GRID 1024
#define HSTRIDE 11 // u64 slots/thread: 22-word stride => LDS bank-conflict-free;
                   // slot 10 is a pad that also absorbs a (impossible) bin==10.

typedef __attribute__((ext_vector_type(4))) float v4f;
typedef __attribute__((ext_vector_type(2))) float v2f;
typedef __attribute__((ext_vector_type(8))) float v8f;

__device__ __forceinline__ unsigned long long pair_pack(float s, unsigned c) {
  return ((unsigned long long)c << 32) | (unsigned long long)__float_as_uint(s);
}
__device__ __forceinline__ float pair_s(unsigned long long p) {
  return __uint_as_float((unsigned)p);
}
__device__ __forceinline__ unsigned pair_c(unsigned long long p) {
  return (unsigned)(p >> 32);
}

__global__ void ghm_init(float* gsum, unsigned* gcnt) {
  int t = threadIdx.x;
  if (t < BINS) { gsum[t] = 0.0f; gcnt[t] = 0u; }
}

__device__ __forceinline__ void ghm_accum(unsigned long long* __restrict__ hrow,
                                          float xs, float ts) {
  float g = fabsf(xs - ts);                    // abs folds as src modifier
  int bin = (int)(g * 9.9999f);                // floor(g*(BINS-1e-4)); g<1 => bin<=9
  float lx  = __logf(xs);                      // v_log_f32 (trans co-exec)
  float l1x = __logf(1.0f - xs);
  // -(t*lx + (1-t)*l1x) = t*(l1x-lx) - l1x  (negation is a free src modifier)
  float bce = fmaf(ts, l1x - lx, -l1x);
  unsigned long long p = hrow[bin];                          // ds_load_b64
  hrow[bin] = pair_pack(pair_s(p) + bce, pair_c(p) + 1u);    // ds_store_b64
}

__global__ __launch_bounds__(TPB) void ghm_main(const float* __restrict__ x,
                                                const float* __restrict__ tgt,
                                                float* __restrict__ gsum,
                                                unsigned* __restrict__ gcnt,
                                                unsigned n) {
  __shared__ unsigned long long hist[TPB * HSTRIDE];  // 22528 B
  const unsigned tid = threadIdx.x;
  unsigned long long* __restrict__ hrow = &hist[tid * HSTRIDE];
#pragma unroll
  for (int b = 0; b < HSTRIDE; ++b) hrow[b] = 0ull;

  const unsigned n4 = n >> 2;                       // fits 32-bit (n from int)
  const unsigned gid = blockIdx.x * TPB + tid;
  const unsigned gstride = gridDim.x * TPB;
  const v4f* x4 = (const v4f*)x;
  const v4f* t4 = (const v4f*)tgt;

  // Streaming pass: 128-bit NT loads, grid-stride, perfectly coalesced.
  for (unsigned i = gid; i < n4; i += gstride) {
    v4f xv = __builtin_nontemporal_load(x4 + i);
    v4f tv = __builtin_nontemporal_load(t4 + i);
#pragma unroll
    for (int c = 0; c < 4; ++c) ghm_accum(hrow, xv[c], tv[c]);
  }
  // Scalar tail (n not multiple of 4)
  for (unsigned i = (n4 << 2) + gid; i < n; i += gstride) {
    float xs = __builtin_nontemporal_load(x + i);
    float ts = __builtin_nontemporal_load(tgt + i);
    ghm_accum(hrow, xs, ts);
  }

  // Workgroup tree reduction over the 256 private histograms.
  __syncthreads();
  for (unsigned off = TPB / 2; off > 0; off >>= 1) {
    if (tid < off) {
#pragma unroll
      for (int b = 0; b < BINS; ++b) {
        unsigned long long a = hrow[b];
        unsigned long long o = hist[(tid + off) * HSTRIDE + b];
        hrow[b] = pair_pack(pair_s(a) + pair_s(o), pair_c(a) + pair_c(o));
      }
    }
    __syncthreads();
  }
  if (tid < BINS) {
    unsigned long long p = hist[tid];  // row 0 holds the WG totals
    atomicAdd(&gsum[tid], pair_s(p));
    atomicAdd(&gcnt[tid], pair_c(p));
  }
}

// One wave32. EXEC is all ones at the WMMA (divergence closed before it).
__global__ __launch_bounds__(32) void ghm_finalize(const float* __restrict__ gsum,
                                                   const unsigned* __restrict__ gcnt,
                                                   float* __restrict__ out,
                                                   unsigned n) {
  __shared__ float sS[BINS];
  __shared__ float sC[BINS];
  const int lane = threadIdx.x;
  if (lane < BINS) { sS[lane] = gsum[lane]; sC[lane] = (float)gcnt[lane]; }
  __syncthreads();

  float ne = 0.0f;
#pragma unroll
  for (int b = 0; b < BINS; ++b) ne += (sC[b] > 0.0f) ? 1.0f : 0.0f;

  float term = 0.0f;
  if (lane < BINS) {
    float gd = fmaxf(sC[lane] * ne, 1.0f);   // clip(count*nonempty, 1)
    term = ((float)n / gd) * sS[lane];       // beta_b * S_b
  }

  // WMMA reduction of the 10 terms: A(16x4) has term in VGPR0 lanes 0..9, rest 0;
  // B = all ones (layout-independent). D row m = sum_k A[m][k] = term[m] for m<10.
  // D layout: lane 0 VGPR r = row r (r=0..7), lane 16 VGPR r = row 8+r.
  v2f a;  a.x = term;  a.y = 0.0f;
  v2f bm; bm.x = 1.0f; bm.y = 1.0f;
  v8f c = {};
  c = __builtin_amdgcn_wmma_f32_16x16x4_f32(false, a, false, bm, (short)0, c,
                                            false, false);
  float partial = c[0] + c[1] + c[2] + c[3] + c[4] + c[5] + c[6] + c[7];
  float hi = __shfl(partial, 16, 32);        // lane16 carries rows 8..9
  if (lane == 0) out[0] = (partial + hi) / (float)n;  // mean
}

extern "C" void kernel_launch(void* const* d_in, const int* in_sizes, int n_in,
                              void* d_out, int out_size, void* d_ws, size_t ws_size,
                              hipStream_t stream) {
  const float* x = (const float*)d_in[0];
  const float* t = (const float*)d_in[1];
  float* out = (float*)d_out;
  const unsigned n = (unsigned)in_sizes[0];

  float* gsum = (float*)d_ws;
  unsigned* gcnt = (unsigned*)((char*)d_ws + BINS * sizeof(float));

  ghm_init<<<1, 32, 0, stream>>>(gsum, gcnt);
  ghm_main<<<GRID, TPB, 0, stream>>>(x, t, gsum, gcnt, n);
  ghm_finalize<<<1, 32, 0, stream>>>(gsum, gcnt, out, n);
}